// CageSkinningV2_76579266888180
// MI455X (gfx1250) — compile-verified
//
#include <hip/hip_runtime.h>
#include <hip/hip_bf16.h>

typedef __attribute__((ext_vector_type(2))) float v2f;
typedef __attribute__((ext_vector_type(8))) float v8f;

#define NPTS   8192
#define NB     16
#define NCAGE  42
#define NFACE  80

// ---------------------------------------------------------------------------
// Phase 0: concat(source_f, target_f) -> feat (16,512)
// ---------------------------------------------------------------------------
__global__ void concat_feat_kernel(const float* __restrict__ sf,
                                   const float* __restrict__ tf,
                                   float* __restrict__ feat) {
    int i = blockIdx.x * blockDim.x + threadIdx.x;
    if (i >= NB * 512) return;
    int m = i >> 9, k = i & 511;
    feat[i] = (k < 256) ? sf[m * 256 + k] : tf[m * 256 + k - 256];
}

// ---------------------------------------------------------------------------
// Phase 1: optimize_cage. One block per (batch, cage-vertex).
// c(t) = s*c0 ; dist^2 = s^2*|c0|^2 + min_p (|p|^2 - 2 s c0.p)
// Per-point (|p|^2, c0.p) precomputed once into registers; each of the 100
// sequential iterations is 1 FMA+min per point plus a wave/LDS reduction.
// ---------------------------------------------------------------------------
__global__ void cage_opt_kernel(const float* __restrict__ src,   // (B,3,N)
                                const float* __restrict__ tmpl,  // (1,3,42)
                                float* __restrict__ cage) {      // (B,3,42)
    const int b = blockIdx.x / NCAGE;
    const int v = blockIdx.x % NCAGE;
    const int tid = threadIdx.x;

    const float cx = tmpl[v], cy = tmpl[NCAGE + v], cz = tmpl[2 * NCAGE + v];
    const float C2 = cx * cx + cy * cy + cz * cz;

    const float* px = src + (size_t)b * 3 * NPTS;
    const float* py = px + NPTS;
    const float* pz = px + 2 * NPTS;

    float aa[32], bd[32];
#pragma unroll
    for (int i = 0; i < 32; ++i) {
        int p = tid + i * 256;
        float x = px[p], y = py[p], z = pz[p];
        aa[i] = x * x + y * y + z * z;
        bd[i] = x * cx + y * cy + z * cz;
    }

    __shared__ float s_sh;
    __shared__ float red[8];
    if (tid == 0) s_sh = 1.0f;
    __syncthreads();

    for (int it = 0; it < 100; ++it) {
        float s = s_sh;
        float m = 3.4e38f;
#pragma unroll
        for (int i = 0; i < 32; ++i)
            m = fminf(m, fmaf(-2.0f * s, bd[i], aa[i]));
#pragma unroll
        for (int off = 16; off > 0; off >>= 1)
            m = fminf(m, __shfl_xor(m, off, 32));
        if ((tid & 31) == 0) red[tid >> 5] = m;
        __syncthreads();
        if (tid == 0) {
            float mm = red[0];
#pragma unroll
            for (int w = 1; w < 8; ++w) mm = fminf(mm, red[w]);
            float d2 = s * s * C2 + mm;        // min squared distance
            if (d2 > 0.16f) s_sh = s * 0.99f;  // min_d > 0.4
        }
        __syncthreads();
    }
    if (tid == 0) {
        float s = s_sh;
        float* cg = cage + (size_t)b * 3 * NCAGE;
        cg[v] = s * cx;
        cg[NCAGE + v] = s * cy;
        cg[2 * NCAGE + v] = s * cz;
    }
}

// ---------------------------------------------------------------------------
// Phase 2: MLP layer via V_WMMA_F32_16X16X4_F32.  M=16 (batch) is exactly one
// WMMA tile; one wave per 16-column tile, loop K in steps of 4.
// A (16xK row-major), W (KxN row-major), out (16xN row-major).
// Partial last tile (N=126): column index is CLAMPED, not predicated — a
// lane's B values only affect that lane's output column, which is never
// stored for invalid lanes. Keeps the inner loop free of exec-mask branches.
// ---------------------------------------------------------------------------
__global__ void mlp_gemm_wmma_kernel(const float* __restrict__ A,
                                     const float* __restrict__ W,
                                     const float* __restrict__ bias,
                                     float* __restrict__ out,
                                     int K, int N, int do_relu) {
    const int lane = threadIdx.x;     // 32 threads = 1 wave
    const int half = lane >> 4;       // 0 | 1
    const int nloc = lane & 15;
    const int col = blockIdx.x * 16 + nloc;
    const bool colOK = (col < N);
    const int colc = colOK ? col : (N - 1);   // clamp (see note above)

    // A 16x4 frag: lanes 0-15 rows M, VGPR0=K(2*half), VGPR1=K(2*half+1)
    const float* Ap = A + nloc * K + 2 * half;
    // B 4x16 frag: row striped across lanes; VGPR v <-> K = v + 2*half
    const float* Wp = W + (2 * half) * N + colc;

    v8f acc = {};
#pragma unroll 4
    for (int k0 = 0; k0 < K; k0 += 4) {
        v2f a, bm;
        a.x = Ap[k0];
        a.y = Ap[k0 + 1];
        bm.x = Wp[k0 * N];
        bm.y = Wp[k0 * N + N];
        acc = __builtin_amdgcn_wmma_f32_16x16x4_f32(
            /*neg_a=*/false, a, /*neg_b=*/false, bm,
            /*c_mod=*/(short)0, acc, /*reuse_a=*/false, /*reuse_b=*/false);
    }

    if (colOK) {
        float bv = bias[col];
#pragma unroll
        for (int v = 0; v < 8; ++v) {
            int row = v + 8 * half;    // C/D: VGPR v, lane-half offset 8
            float val = acc[v] + bv;
            if (do_relu) val = fmaxf(val, 0.0f);
            out[row * N + col] = val;
        }
    }
}

// ---------------------------------------------------------------------------
// Phase 3: new_cage = cage + offset ; emit cage_t / new_cage_t / infl outputs
// d_out layout (floats): cage_t @0 (2016) | new_cage_t @2016 | deformed @4032
// (393216) | weights @397248 (5505024) | influence_offset @5902272 (2016)
// ---------------------------------------------------------------------------
__global__ void cage_update_kernel(const float* __restrict__ cage,
                                   const float* __restrict__ infl,
                                   float* __restrict__ new_cage,
                                   float* __restrict__ out) {
    int i = blockIdx.x * blockDim.x + threadIdx.x;
    if (i >= NB * 3 * NCAGE) return;
    int b = i / 126;
    int j = i - b * 126;       // j = d*42 + v
    int d = j / NCAGE, v = j - d * NCAGE;
    float c = cage[i];
    float off = infl[i];
    float nc = c + off;
    new_cage[i] = nc;
    out[(size_t)b * 126 + v * 3 + d] = c;                 // cage_t
    out[2016 + (size_t)b * 126 + v * 3 + d] = nc;         // new_cage_t
    out[5902272 + i] = off;                               // influence_offset
}

// ---------------------------------------------------------------------------
// Phase 4: mean-value coordinates + deform. 128 points per block.
// Per-thread scatter accumulators in LDS, laid out [vertex][thread]
// (face indices are wave-uniform -> conflict-free, lane==bank).
// ---------------------------------------------------------------------------
__global__ void __launch_bounds__(128)
mvc_kernel(const float* __restrict__ src,        // (B,3,N) query source
           const int* __restrict__ faces,        // (80,3)
           const float* __restrict__ cage,       // (B,3,42)
           const float* __restrict__ new_cage,   // (B,3,42)
           float* __restrict__ out) {
    __shared__ float vx[NCAGE], vy[NCAGE], vz[NCAGE];
    __shared__ float nvx[NCAGE], nvy[NCAGE], nvz[NCAGE];
    __shared__ int   fidx[NFACE * 3];
    __shared__ float wjN[NCAGE * 128];
    __shared__ float wjI[NCAGE * 128];

    const int tid = threadIdx.x;
    const int b = blockIdx.x >> 6;                 // 64 blocks per batch
    const int p0 = (blockIdx.x & 63) << 7;         // first point of block
    const int p = p0 + tid;                        // this thread's point

    const float* cg = cage + (size_t)b * 126;
    const float* ncg = new_cage + (size_t)b * 126;
    if (tid < NCAGE) {
        vx[tid] = cg[tid]; vy[tid] = cg[NCAGE + tid]; vz[tid] = cg[2 * NCAGE + tid];
        nvx[tid] = ncg[tid]; nvy[tid] = ncg[NCAGE + tid]; nvz[tid] = ncg[2 * NCAGE + tid];
    }
    for (int i = tid; i < NFACE * 3; i += 128) fidx[i] = faces[i];
#pragma unroll 1
    for (int v = 0; v < NCAGE; ++v) { wjN[v * 128 + tid] = 0.f; wjI[v * 128 + tid] = 0.f; }
    __syncthreads();

    const float* sb = src + (size_t)b * 3 * NPTS;
    const float qx = sb[p], qy = sb[NPTS + p], qz = sb[2 * NPTS + p];

    // close detection (dj < 1e-8)
    unsigned long long closeMask = 0ull;
    bool anyClose = false;
#pragma unroll 1
    for (int c = 0; c < NCAGE; ++c) {
        float ux = vx[c] - qx, uy = vy[c] - qy, uz = vz[c] - qz;
        float d = sqrtf(ux * ux + uy * uy + uz * uz);
        if (d < 1e-8f) { closeMask |= (1ull << c); anyClose = true; }
    }

    bool anyInside = false;
    const float lclamp = 2.0f - 2e-5f;
    const float cclamp = 1.0f - 1e-5f;

#pragma unroll 1
    for (int f = 0; f < NFACE; ++f) {
        const int i0 = fidx[f * 3], i1 = fidx[f * 3 + 1], i2 = fidx[f * 3 + 2];

        float u0x = vx[i0] - qx, u0y = vy[i0] - qy, u0z = vz[i0] - qz;
        float u1x = vx[i1] - qx, u1y = vy[i1] - qy, u1z = vz[i1] - qz;
        float u2x = vx[i2] - qx, u2y = vy[i2] - qy, u2z = vz[i2] - qz;
        float d0 = sqrtf(u0x * u0x + u0y * u0y + u0z * u0z);
        float d1 = sqrtf(u1x * u1x + u1y * u1y + u1z * u1z);
        float d2 = sqrtf(u2x * u2x + u2y * u2y + u2z * u2z);
        float r0 = 1.0f / d0, r1 = 1.0f / d1, r2 = 1.0f / d2;
        u0x *= r0; u0y *= r0; u0z *= r0;
        u1x *= r1; u1y *= r1; u1z *= r1;
        u2x *= r2; u2y *= r2; u2z *= r2;

        // li[k] = ||u[r1[k]] - u[r2[k]]||, r1=[1,2,0], r2=[2,0,1]
        float ax, ay, az;
        ax = u1x - u2x; ay = u1y - u2y; az = u1z - u2z;
        float l0 = sqrtf(ax * ax + ay * ay + az * az);
        ax = u2x - u0x; ay = u2y - u0y; az = u2z - u0z;
        float l1 = sqrtf(ax * ax + ay * ay + az * az);
        ax = u0x - u1x; ay = u0y - u1y; az = u0z - u1z;
        float l2 = sqrtf(ax * ax + ay * ay + az * az);
        l0 = fminf(l0, lclamp); l1 = fminf(l1, lclamp); l2 = fminf(l2, lclamp);

        float t0 = 2.0f * asinf(0.5f * l0);
        float t1 = 2.0f * asinf(0.5f * l1);
        float t2 = 2.0f * asinf(0.5f * l2);
        // exact: sin(2 asin(l/2)) = l*sqrt(1-l^2/4), cos = 1-l^2/2
        float st0 = l0 * sqrtf(fmaxf(0.f, 1.0f - 0.25f * l0 * l0));
        float st1 = l1 * sqrtf(fmaxf(0.f, 1.0f - 0.25f * l1 * l1));
        float st2 = l2 * sqrtf(fmaxf(0.f, 1.0f - 0.25f * l2 * l2));
        float ct0 = 1.0f - 0.5f * l0 * l0;
        float ct1 = 1.0f - 0.5f * l1 * l1;
        float ct2 = 1.0f - 0.5f * l2 * l2;

        float h = 0.5f * (t0 + t1 + t2);
        float sh = sinf(h), ch = cosf(h);

        // ci[k] = 2 sin(h) sin(h-tk) / (st[r1k] st[r2k]) - 1
        float c0 = 2.0f * sh * (sh * ct0 - ch * st0) / (st1 * st2) - 1.0f;
        float c1 = 2.0f * sh * (sh * ct1 - ch * st1) / (st2 * st0) - 1.0f;
        float c2 = 2.0f * sh * (sh * ct2 - ch * st2) / (st0 * st1) - 1.0f;
        c0 = fminf(fmaxf(c0, -cclamp), cclamp);
        c1 = fminf(fmaxf(c1, -cclamp), cclamp);
        c2 = fminf(fmaxf(c2, -cclamp), cclamp);

        float det = u0x * (u1y * u2z - u1z * u2y)
                  - u0y * (u1x * u2z - u1z * u2x)
                  + u0z * (u1x * u2y - u1y * u2x);
        float sg = (det > 0.f) ? 1.f : ((det < 0.f) ? -1.f : 0.f);
        float s0 = sg * sqrtf(1.0f - c0 * c0);
        float s1 = sg * sqrtf(1.0f - c1 * c1);
        float s2 = sg * sqrtf(1.0f - c2 * c2);

        float w0 = (t0 - c1 * t2 - c2 * t1) / (d0 * st1 * s2);
        float w1 = (t1 - c2 * t0 - c0 * t2) / (d1 * st2 * s0);
        float w2 = (t2 - c0 * t1 - c1 * t0) / (d2 * st0 * s1);

        wjN[i0 * 128 + tid] += w0;
        wjN[i1 * 128 + tid] += w1;
        wjN[i2 * 128 + tid] += w2;

        if (3.14159265358979f - h < 1e-4f) {   // inside
            anyInside = true;
            wjI[i0 * 128 + tid] += st0 * d2 * d1;
            wjI[i1 * 128 + tid] += st1 * d0 * d2;
            wjI[i2 * 128 + tid] += st2 * d1 * d0;
        }
    }

    // select accumulator, apply close override, compute sum
    float sum = 0.0f;
#pragma unroll 1
    for (int c = 0; c < NCAGE; ++c) {
        float w = anyInside ? wjI[c * 128 + tid] : wjN[c * 128 + tid];
        if (anyClose) w = ((closeMask >> c) & 1ull) ? 1.0f : 0.0f;
        wjN[c * 128 + tid] = w;
        sum += w;
    }
    if (sum == 0.0f) sum = 1.0f;
    const float inv = 1.0f / sum;

    // normalize in LDS + deform with this thread's own column
    float dx = 0.f, dy = 0.f, dz = 0.f;
#pragma unroll 1
    for (int c = 0; c < NCAGE; ++c) {
        float w = wjN[c * 128 + tid] * inv;
        wjN[c * 128 + tid] = w;
        dx += w * nvx[c]; dy += w * nvy[c]; dz += w * nvz[c];
    }
    float* dout = out + 4032 + (size_t)b * 3 * NPTS;   // deformed (B,3,P)
    dout[p] = dx;
    dout[NPTS + p] = dy;
    dout[2 * NPTS + p] = dz;

    __syncthreads();
    // coalesced cooperative write of the block's contiguous weights region:
    // rows [p0, p0+128) of the (B*P, 42) weights array = 5376 consecutive floats
    float* wbase = out + 397248 + ((size_t)b * NPTS + p0) * NCAGE;
#pragma unroll 1
    for (int i = tid; i < 128 * NCAGE; i += 128) {
        int row = i / NCAGE;
        int c = i - row * NCAGE;
        wbase[i] = wjN[c * 128 + row];
    }
}

// ---------------------------------------------------------------------------
extern "C" void kernel_launch(void* const* d_in, const int* in_sizes, int n_in,
                              void* d_out, int out_size, void* d_ws, size_t ws_size,
                              hipStream_t stream) {
    const float* src_shape = (const float*)d_in[0];   // (16,3,8192)
    // d_in[1] target_shape: unused by outputs
    const float* src_f = (const float*)d_in[2];       // (16,256)
    const float* tgt_f = (const float*)d_in[3];       // (16,256)
    const float* tmpl  = (const float*)d_in[4];       // (1,3,42)
    const int*   faces = (const int*)d_in[5];         // (80,3)
    const float* W1 = (const float*)d_in[6];   const float* b1 = (const float*)d_in[7];
    const float* W2 = (const float*)d_in[8];   const float* b2 = (const float*)d_in[9];
    const float* W3 = (const float*)d_in[10];  const float* b3 = (const float*)d_in[11];
    const float* W4 = (const float*)d_in[12];  const float* b4 = (const float*)d_in[13];
    float* out = (float*)d_out;

    // workspace layout (floats)
    float* ws = (float*)d_ws;
    float* cage     = ws;            // 2016
    float* feat     = ws + 2016;     // 8192
    float* h1       = ws + 10208;    // 8192
    float* h2       = ws + 18400;    // 8192
    float* h3       = ws + 26592;    // 4096
    float* infl     = ws + 30688;    // 2016
    float* new_cage = ws + 32704;    // 2016

    // Phase 0 + 1 (independent)
    concat_feat_kernel<<<(NB * 512 + 255) / 256, 256, 0, stream>>>(src_f, tgt_f, feat);
    cage_opt_kernel<<<NB * NCAGE, 256, 0, stream>>>(src_shape, tmpl, cage);

    // Phase 2: MLP, fp32 WMMA 16x16x4
    mlp_gemm_wmma_kernel<<<512 / 16, 32, 0, stream>>>(feat, W1, b1, h1, 512, 512, 1);
    mlp_gemm_wmma_kernel<<<512 / 16, 32, 0, stream>>>(h1, W2, b2, h2, 512, 512, 1);
    mlp_gemm_wmma_kernel<<<256 / 16, 32, 0, stream>>>(h2, W3, b3, h3, 512, 256, 1);
    mlp_gemm_wmma_kernel<<<(126 + 15) / 16, 32, 0, stream>>>(h3, W4, b4, infl, 256, 126, 0);

    // Phase 3
    cage_update_kernel<<<(NB * 126 + 255) / 256, 256, 0, stream>>>(cage, infl, new_cage, out);

    // Phase 4
    mvc_kernel<<<NB * (NPTS / 128), 128, 0, stream>>>(src_shape, faces, cage, new_cage, out);
}